// VectorQuantizer_50182397886711
// MI455X (gfx1250) — compile-verified
//
#include <hip/hip_runtime.h>
#include <hip/hip_bf16.h>
#include <stdint.h>

typedef __attribute__((ext_vector_type(2))) float v2f;
typedef __attribute__((ext_vector_type(8))) float v8f;

#define NUM_EMB    1024
#define EMB_DIM    64
#define N_VEC      65536            // 64*32*32
#define Q_ELEMS    4194304          // N_VEC * EMB_DIM
#define KCHUNK     128              // embeddings staged in LDS per iteration
#define LDS_STRIDE 68               // 64 + 4 pad floats -> conflict-free b64 reads
#define WAVES_PB   8                // 256 threads
#define ROWS_PB    (WAVES_PB * 16)  // 128 X-rows per block

// ---------------------------------------------------------------------------
// Kernel A: embedding norms ||e_k||^2 -> workspace; zero the loss accumulator.
// ---------------------------------------------------------------------------
__global__ void vq_prep_kernel(const float* __restrict__ E,
                               float* __restrict__ enorm,
                               float* __restrict__ loss_slot) {
    int k = blockIdx.x * blockDim.x + threadIdx.x;
    if (k == 0) *loss_slot = 0.0f;
    if (k < NUM_EMB) {
        const float* e = E + (size_t)k * EMB_DIM;
        float s = 0.0f;
#pragma unroll
        for (int d = 0; d < EMB_DIM; d += 4) {
            float4 v = *(const float4*)(e + d);
            s += v.x * v.x + v.y * v.y + v.z * v.z + v.w * v.w;
        }
        enorm[k] = s;
    }
}

// ---------------------------------------------------------------------------
// Kernel B: WMMA distance GEMM + argmin + gather + loss.
// Each wave: 16 rows of X vs all 1024 embeddings.
// E chunks staged global->LDS with CDNA5 async-to-LDS DMA (ASYNCcnt tracked).
// ---------------------------------------------------------------------------
__global__ __launch_bounds__(256) void vq_main_kernel(
    const float* __restrict__ X,
    const float* __restrict__ E,
    const float* __restrict__ enorm,
    float* __restrict__ qout,
    float* __restrict__ loss_slot,
    float* __restrict__ idxout) {

    __shared__ float shE[KCHUNK * LDS_STRIDE];
    __shared__ int   shIdx[WAVES_PB * 16];

    const int tid   = threadIdx.x;
    const int wave  = tid >> 5;       // 0..7
    const int lane  = tid & 31;
    const int lhalf = lane & 15;      // 0..15 within half-wave
    const int hi    = lane >> 4;      // 0: K=0,1 ; 1: K=2,3 (f32 A/B frag layout)

    const int rowBase = (blockIdx.x * WAVES_PB + wave) * 16;
    const int mrow    = rowBase + lhalf;   // this lane's A-matrix row

    // ---- hoist A fragments: X[mrow, :] split into 16 chunks of K=4 ----
    // ISA 16x4 f32 A layout: lanes 0-15 hold {K=0,K=1}, lanes 16-31 hold {K=2,K=3}
    v2f a[16];
    {
        const float* xr = X + (size_t)mrow * EMB_DIM + hi * 2;
#pragma unroll
        for (int c = 0; c < 16; ++c)
            a[c] = *(const v2f*)(xr + 4 * c);
    }

    // per-lane running argmin: 8 accumulator rows per lane (C layout: VGPR r ->
    // row r (lanes 0-15) / row r+8 (lanes 16-31), column = lane%16)
    float best[8];
    int   bidx[8];
#pragma unroll
    for (int r = 0; r < 8; ++r) { best[r] = 3.4e38f; bidx[r] = 0; }

    // staging geometry: this thread copies 8 b128 slots per chunk
    // slot idx = i*256 + tid ; row = idx>>4 (0..127), c4 = idx&15
    for (int kc = 0; kc < NUM_EMB; kc += KCHUNK) {
        __syncthreads();  // all waves done reading previous chunk (WAR)

        // ---- async DMA: E[kc .. kc+127][0..63] -> LDS, no VGPR round trip ----
#pragma unroll
        for (int i = 0; i < 8; ++i) {
            const int idx = i * 256 + tid;
            const int r   = idx >> 4;
            const int c4  = idx & 15;
            const uint32_t goff = (uint32_t)((kc + r) * (EMB_DIM * 4) + c4 * 16);
            const uint32_t loff = (uint32_t)(uintptr_t)(&shE[r * LDS_STRIDE + c4 * 4]);
            asm volatile("global_load_async_to_lds_b128 %0, %1, %2"
                         :
                         : "v"(loff), "v"(goff), "s"(E)
                         : "memory");
        }
        asm volatile("s_wait_asynccnt 0x0" ::: "memory");
        __syncthreads();  // everyone's DMA landed in LDS

        // ---- 8 tiles of 16 embeddings, interleaved 2 at a time for ILP ----
#pragma unroll
        for (int t = 0; t < 8; t += 2) {
            const float* brow0 = &shE[((t + 0) * 16 + lhalf) * LDS_STRIDE + hi * 2];
            const float* brow1 = &shE[((t + 1) * 16 + lhalf) * LDS_STRIDE + hi * 2];

            v8f acc0 = {};
            v8f acc1 = {};
#pragma unroll
            for (int c = 0; c < 16; ++c) {
                v2f b0 = *(const v2f*)(brow0 + 4 * c);
                v2f b1 = *(const v2f*)(brow1 + 4 * c);
                acc0 = __builtin_amdgcn_wmma_f32_16x16x4_f32(
                    false, a[c], false, b0, (short)0, acc0, false, false);
                acc1 = __builtin_amdgcn_wmma_f32_16x16x4_f32(
                    false, a[c], false, b1, (short)0, acc1, false, false);
            }

            // dist(m,k) - ||x_m||^2 = ||e_k||^2 - 2 * S(m,k)  (row-const dropped)
            const int   col0 = kc + t * 16 + lhalf;
            const int   col1 = col0 + 16;
            const float en0  = enorm[col0];
            const float en1  = enorm[col1];
#pragma unroll
            for (int r = 0; r < 8; ++r) {
                const float d0 = __builtin_fmaf(-2.0f, acc0[r], en0);
                if (d0 < best[r]) { best[r] = d0; bidx[r] = col0; }
                const float d1 = __builtin_fmaf(-2.0f, acc1[r], en1);
                if (d1 < best[r]) { best[r] = d1; bidx[r] = col1; }
            }
        }
    }

    // ---- per-row argmin reduction across the 16 lanes of each half-wave ----
#pragma unroll
    for (int r = 0; r < 8; ++r) {
        float bv = best[r];
        int   bi = bidx[r];
#pragma unroll
        for (int msk = 8; msk >= 1; msk >>= 1) {
            float ov = __shfl_xor(bv, msk, 32);
            int   oi = __shfl_xor(bi, msk, 32);
            if (ov < bv || (ov == bv && oi < bi)) { bv = ov; bi = oi; }
        }
        if (lhalf == 0)                       // lane 0 -> rows 0..7, lane 16 -> rows 8..15
            shIdx[wave * 16 + hi * 8 + r] = bi;
    }
    // same-wave LDS RAW: compiler inserts s_wait_dscnt; no workgroup barrier needed.

    // ---- gather quantized rows, accumulate commitment-loss partial ----
    float lsum = 0.0f;
#pragma unroll
    for (int rr = 0; rr < 16; ++rr) {
        const int    k   = shIdx[wave * 16 + rr];
        const int    row = rowBase + rr;
        const float2 e2  = *(const float2*)(E + (size_t)k  * EMB_DIM + lane * 2);
        const float2 x2  = *(const float2*)(X + (size_t)row * EMB_DIM + lane * 2);
        const float d0 = e2.x - x2.x;
        const float d1 = e2.y - x2.y;
        lsum += d0 * d0 + d1 * d1;
        *(float2*)(qout + (size_t)row * EMB_DIM + lane * 2) = e2;
        if (lane == 0) idxout[row] = (float)k;
    }
#pragma unroll
    for (int msk = 16; msk >= 1; msk >>= 1)
        lsum += __shfl_xor(lsum, msk, 32);
    if (lane == 0)
        atomicAdd(loss_slot, lsum * (0.25f / (float)Q_ELEMS));
}

// ---------------------------------------------------------------------------
extern "C" void kernel_launch(void* const* d_in, const int* in_sizes, int n_in,
                              void* d_out, int out_size, void* d_ws, size_t ws_size,
                              hipStream_t stream) {
    const float* X = (const float*)d_in[0];   // [64,32,32,64] f32
    const float* E = (const float*)d_in[1];   // [1024,64] f32

    float* out    = (float*)d_out;
    float* qout   = out;                 // [Q_ELEMS]
    float* loss   = out + Q_ELEMS;       // [1]
    float* idxout = out + Q_ELEMS + 1;   // [N_VEC] (indices as float)

    float* enorm = (float*)d_ws;         // [NUM_EMB] floats

    vq_prep_kernel<<<dim3((NUM_EMB + 255) / 256), dim3(256), 0, stream>>>(E, enorm, loss);
    vq_main_kernel<<<dim3(N_VEC / ROWS_PB), dim3(256), 0, stream>>>(
        X, E, enorm, qout, loss, idxout);
}